// CTCConsistencyLoss_7310034338203
// MI455X (gfx1250) — compile-verified
//
#include <hip/hip_runtime.h>
#include <hip/hip_bf16.h>
#include <math.h>

// Problem constants (match reference)
#define Bn 32
#define Tn 1000
#define Dn 768
#define Vn 31
#define Vp 32      // padded vocab
#define Ln 200
#define Sn 401     // 2*L+1
#define NEGF (-1e30f)

typedef __attribute__((ext_vector_type(2))) float v2f;
typedef __attribute__((ext_vector_type(8))) float v8f;

// ---------------------------------------------------------------------------
// Kernel 0: zero the scalar output (graph-safe init)
// ---------------------------------------------------------------------------
__global__ void ctc_zero_out(float* out) {
    if (threadIdx.x == 0 && blockIdx.x == 0) out[0] = 0.0f;
}

// ---------------------------------------------------------------------------
// Kernel 1: pad W (31x768) -> Wpad (32x768) with a zero row so the second
// N-tile of the WMMA GEMM needs no per-lane predication (EXEC must be all-1s).
// ---------------------------------------------------------------------------
__global__ void ctc_pad_w(const float* __restrict__ W, float* __restrict__ Wpad) {
    int i = blockIdx.x * blockDim.x + threadIdx.x;
    if (i < Vp * Dn) {
        int r = i / Dn;
        Wpad[i] = (r < Vn) ? W[i] : 0.0f;
    }
}

// ---------------------------------------------------------------------------
// Kernel 2: logits = pred @ W^T + b, then log_softmax over V, via
// V_WMMA_F32_16X16X4_F32. 256 threads = 8 waves; each wave computes a 32x32
// output tile (2 M-tiles x 2 N-tiles), K-loop of 192 WMMA steps per tile pair.
// ---------------------------------------------------------------------------
__global__ void __launch_bounds__(256)
ctc_gemm_logsoftmax(const float* __restrict__ pred,
                    const float* __restrict__ Wpad,
                    const float* __restrict__ bias,
                    float* __restrict__ logp) {
    __shared__ float stage[8][32][33];   // per-wave 32x32 tile, padded stride

    const int lane = threadIdx.x & 31;
    const int wave = threadIdx.x >> 5;
    const int rowBase = (blockIdx.x * 8 + wave) * 32;   // 125*8*32 = 32000 rows

    // A-operand layout (16x4 f32): lane l -> M = l&15, VGPR v holds K = 2*(l>>4)+v
    const int m    = lane & 15;
    const int koff = (lane >> 4) * 2;

    const float* a0p = pred + (size_t)(rowBase + m) * Dn + koff;       // M-tile 0
    const float* a1p = a0p + (size_t)16 * Dn;                          // M-tile 1
    // B-operand (4x16 = W^T tile): lane l -> N = l&15, VGPR v holds K = 2*(l>>4)+v
    const float* b0p = Wpad + (size_t)m * Dn + koff;                   // N-tile 0 (cols 0..15)
    const float* b1p = Wpad + (size_t)(m + 16) * Dn + koff;            // N-tile 1 (cols 16..31)

    v8f c00 = {}, c01 = {}, c10 = {}, c11 = {};

#pragma unroll 4
    for (int k = 0; k < Dn; k += 4) {
        v2f a0 = *(const v2f*)(a0p + k);
        v2f a1 = *(const v2f*)(a1p + k);
        v2f b0 = *(const v2f*)(b0p + k);
        v2f b1 = *(const v2f*)(b1p + k);
        // (neg_a, A, neg_b, B, c_mod, C, reuse_a, reuse_b)
        c00 = __builtin_amdgcn_wmma_f32_16x16x4_f32(false, a0, false, b0, (short)0, c00, false, false);
        c01 = __builtin_amdgcn_wmma_f32_16x16x4_f32(false, a0, false, b1, (short)0, c01, false, false);
        c10 = __builtin_amdgcn_wmma_f32_16x16x4_f32(false, a1, false, b0, (short)0, c10, false, false);
        c11 = __builtin_amdgcn_wmma_f32_16x16x4_f32(false, a1, false, b1, (short)0, c11, false, false);
    }

    // C/D layout: VGPR v, lanes 0-15 -> M=v, lanes 16-31 -> M=v+8; N = lane&15
    const int rhalf = (lane >> 4) * 8;
    const int col   = lane & 15;
#pragma unroll
    for (int v = 0; v < 8; ++v) {
        stage[wave][rhalf + v][col]           = c00[v];
        stage[wave][rhalf + v][col + 16]      = c01[v];
        stage[wave][16 + rhalf + v][col]      = c10[v];
        stage[wave][16 + rhalf + v][col + 16] = c11[v];
    }
    __syncthreads();

    // log-softmax over the 31 valid columns; each lane owns one row
    const float* rowv = stage[wave][lane];
    float mx = NEGF;
#pragma unroll
    for (int v = 0; v < Vn; ++v) {
        float x = rowv[v] + bias[v];
        mx = fmaxf(mx, x);
    }
    float sum = 0.0f;
#pragma unroll
    for (int v = 0; v < Vn; ++v) {
        sum += __expf(rowv[v] + bias[v] - mx);
    }
    float lse = mx + __logf(sum);

    float* outp = logp + (size_t)(rowBase + lane) * Vp;
#pragma unroll
    for (int v = 0; v < Vn; ++v) {
        outp[v] = rowv[v] + bias[v] - lse;
    }
    outp[Vn] = NEGF;   // padded column, never gathered
}

// ---------------------------------------------------------------------------
// Kernel 3: CTC forward recursion. One workgroup per batch element; alpha
// double-buffered in LDS; T=1000 serial steps; atomicAdd of nll/tl/B.
// ---------------------------------------------------------------------------
__global__ void __launch_bounds__(256)
ctc_forward(const float* __restrict__ logp,
            const int* __restrict__ targets,
            const int* __restrict__ in_len,
            const int* __restrict__ tgt_len,
            float* __restrict__ out) {
    const int b   = blockIdx.x;
    const int tid = threadIdx.x;

    __shared__ int   ext[Sn];
    __shared__ unsigned char skipv[Sn];
    __shared__ float alpha[2][Sn];
    __shared__ float lprow[Vp];

    // extended label sequence: blank, l1, blank, l2, ..., blank
    for (int s = tid; s < Sn; s += 256) {
        ext[s] = (s & 1) ? targets[b * Ln + (s >> 1)] : 0;
    }
    __syncthreads();
    for (int s = tid; s < Sn; s += 256) {
        int e = ext[s];
        skipv[s] = (unsigned char)((s >= 2) && (e != 0) && (e != ext[s - 2]));
    }

    const float* lpb = logp + (size_t)b * Tn * Vp;

    // t = 0 init
    for (int s = tid; s < Sn; s += 256) {
        float a = NEGF;
        if (s == 0) a = lpb[0];          // blank
        if (s == 1) a = lpb[ext[1]];
        alpha[0][s] = a;
    }
    const int til = in_len[b];
    __syncthreads();

    for (int t = 1; t < Tn; ++t) {
        if (tid < Vp) lprow[tid] = lpb[(size_t)t * Vp + tid];
        __syncthreads();

        const float* ap = alpha[(t - 1) & 1];
        float*       an = alpha[t & 1];
        const bool frozen = (t >= til);

        for (int s = tid; s < Sn; s += 256) {
            float a0 = ap[s];
            float a1 = (s >= 1) ? ap[s - 1] : NEGF;
            float a2 = (s >= 2 && skipv[s]) ? ap[s - 2] : NEGF;
            float mm = fmaxf(a0, fmaxf(a1, a2));
            float nv = mm + __logf(__expf(a0 - mm) + __expf(a1 - mm) + __expf(a2 - mm))
                          + lprow[ext[s]];
            an[s] = frozen ? a0 : nv;
        }
        __syncthreads();
    }

    if (tid == 0) {
        const float* af = alpha[(Tn - 1) & 1];
        int tl  = tgt_len[b];
        float x0 = af[2 * tl - 1];
        float x1 = af[2 * tl];
        float mm = fmaxf(x0, x1);
        float nll = -(mm + __logf(__expf(x0 - mm) + __expf(x1 - mm)));
        if (nll > 1e29f) nll = 0.0f;     // zero_infinity
        atomicAdd(out, nll / (float)tl / (float)Bn);
    }
}

// ---------------------------------------------------------------------------
// Launcher
// ---------------------------------------------------------------------------
extern "C" void kernel_launch(void* const* d_in, const int* in_sizes, int n_in,
                              void* d_out, int out_size, void* d_ws, size_t ws_size,
                              hipStream_t stream) {
    const float* pred    = (const float*)d_in[0];   // (B,T,D) f32
    const int*   targets = (const int*)d_in[1];     // (B*L,) i32
    const int*   ilen    = (const int*)d_in[2];     // (B,) i32
    const int*   tlen    = (const int*)d_in[3];     // (B,) i32
    const float* W       = (const float*)d_in[4];   // (V,D) f32
    const float* bias    = (const float*)d_in[5];   // (V,) f32
    float*       out     = (float*)d_out;

    char*  ws   = (char*)d_ws;
    float* Wpad = (float*)ws;                                    // 32*768 f32
    float* logp = (float*)(ws + (size_t)Vp * Dn * sizeof(float)); // 32000*32 f32

    ctc_zero_out<<<1, 32, 0, stream>>>(out);
    ctc_pad_w<<<(Vp * Dn + 255) / 256, 256, 0, stream>>>(W, Wpad);

    // 32000 rows / (8 waves * 32 rows) = 125 workgroups
    ctc_gemm_logsoftmax<<<125, 256, 0, stream>>>(pred, Wpad, bias, logp);

    ctc_forward<<<Bn, 256, 0, stream>>>(logp, targets, ilen, tlen, out);
}